// DPN_88965952569844
// MI455X (gfx1250) — compile-verified
//
#include <hip/hip_runtime.h>
#include <stdint.h>

// Problem sizes (match reference)
#define B_SZ   4096
#define D_IN   4096
#define HID    16384
#define OUTN   1000
#define K2TOT  (D_IN + HID)   // 20480

// Tiling
#define BM   128
#define BN   128
#define BK   32
#define LDT  36               // LDS row stride in floats: 144B -> 16B-aligned rows, conflict-free frags

typedef float v2f __attribute__((ext_vector_type(2)));
typedef float v8f __attribute__((ext_vector_type(8)));

// WMMA f32 16x16x4 fragment layout (wave32, per CDNA5 ISA 7.12.2):
//  A (16x4): lane L -> M = L%16, holds K = {2*(L/16), 2*(L/16)+1}   (v2f)
//  B (4x16): lane L -> N = L%16, holds K = {2*(L/16), 2*(L/16)+1}   (v2f)
//  C/D (16x16): vgpr r, lane L -> M = r + 8*(L/16), N = L%16        (v8f)

__device__ __forceinline__ v8f wmma_f32(v2f a, v2f b, v8f c) {
  return __builtin_amdgcn_wmma_f32_16x16x4_f32(
      /*neg_a=*/false, a, /*neg_b=*/false, b,
      /*c_mod=*/(short)0, c, /*reuse_a=*/false, /*reuse_b=*/false);
}

// Async global->LDS copy (CDNA5 GLOBAL_LOAD_ASYNC_TO_LDS_B128, GVS mode).
// ldsoff: byte offset within workgroup LDS (generic ptr low 32 bits),
// byteoff: per-lane unsigned byte offset, base: uniform 64-bit SGPR address.
__device__ __forceinline__ void async_ld_b128(uint32_t ldsoff, const float* base,
                                              uint32_t byteoff) {
  asm volatile("global_load_async_to_lds_b128 %0, %1, %2"
               :: "v"(ldsoff), "v"(byteoff),
                  "s"((unsigned long long)(uintptr_t)base)
               : "memory");
}

__device__ __forceinline__ void wait_async0() {
#if __has_builtin(__builtin_amdgcn_s_wait_asynccnt)
  __builtin_amdgcn_s_wait_asynccnt(0);
#else
  asm volatile("s_wait_asynccnt 0x0" ::: "memory");
#endif
}

__device__ __forceinline__ uint32_t lds_u32(const void* p) {
  return (uint32_t)(uintptr_t)p;   // addrspace(3)->flat keeps LDS offset in addr[31:0]
}

// ---------------------------------------------------------------------------
// Kernel 1: h1[B_SZ][HID] = relu(x @ W1^T + b1)
// grid = (HID/BN, B_SZ/BM), block = 256 (8 wave32)
// ---------------------------------------------------------------------------
__global__ __launch_bounds__(256) void DPN_gemm1_relu(
    const float* __restrict__ x, const float* __restrict__ W1,
    const float* __restrict__ b1, float* __restrict__ h1)
{
  __shared__ __align__(16) float As[2][BM * LDT];
  __shared__ __align__(16) float Bs[2][BN * LDT];

  const int tid   = threadIdx.x;
  const int lane  = tid & 31;
  const int wave  = tid >> 5;           // 0..7
  const int lm    = lane & 15;          // row/col within 16
  const int kh    = (lane >> 4) << 1;   // 0 or 2 : K pair offset
  const int waveM = (wave >> 1) * 32;   // 0,32,64,96
  const int waveN = (wave & 1) * 64;    // 0,64

  const int M0 = blockIdx.y * BM;
  const int N0 = blockIdx.x * BN;

  const v8f vzero = {0.f,0.f,0.f,0.f,0.f,0.f,0.f,0.f};
  v8f acc[2][4];
#pragma unroll
  for (int i = 0; i < 2; ++i)
#pragma unroll
    for (int j = 0; j < 4; ++j) acc[i][j] = vzero;

  // Issue async loads of one 128x32 A-tile + 128x32 B-tile into buffer b.
  auto issue = [&](int kk, int b) {
#pragma unroll
    for (int j = 0; j < 4; ++j) {
      const int idx = tid + j * 256;    // 1024 float4 per tile
      const int row = idx >> 3;
      const int c4  = (idx & 7) << 2;
      async_ld_b128(lds_u32(&As[b][row * LDT + c4]), x,
                    (uint32_t)(((M0 + row) * D_IN + kk + c4) * 4));
      async_ld_b128(lds_u32(&Bs[b][row * LDT + c4]), W1,
                    (uint32_t)(((N0 + row) * D_IN + kk + c4) * 4));
    }
  };

  issue(0, 0);
  wait_async0();
  __syncthreads();

  int buf = 0;
  for (int kk = 0; kk < D_IN; kk += BK) {
    const int nbuf = buf ^ 1;
    if (kk + BK < D_IN) issue(kk + BK, nbuf);   // prefetch next tile into other buffer

#pragma unroll
    for (int k4 = 0; k4 < BK; k4 += 4) {
      v2f afrag[2], bfrag[4];
#pragma unroll
      for (int ms = 0; ms < 2; ++ms)
        afrag[ms] = *(const v2f*)(&As[buf][(waveM + ms * 16 + lm) * LDT + k4 + kh]);
#pragma unroll
      for (int ns = 0; ns < 4; ++ns)
        bfrag[ns] = *(const v2f*)(&Bs[buf][(waveN + ns * 16 + lm) * LDT + k4 + kh]);
#pragma unroll
      for (int ms = 0; ms < 2; ++ms)
#pragma unroll
        for (int ns = 0; ns < 4; ++ns)
          acc[ms][ns] = wmma_f32(afrag[ms], bfrag[ns], acc[ms][ns]);
    }

    wait_async0();        // this wave's async copies into nbuf are done
    __syncthreads();      // all waves' copies are done; buf reads all consumed
    buf = nbuf;
  }

  // Epilogue: bias + ReLU, store h1.
#pragma unroll
  for (int ns = 0; ns < 4; ++ns) {
    const int gn   = N0 + waveN + ns * 16 + lm;
    const float bv = b1[gn];
#pragma unroll
    for (int ms = 0; ms < 2; ++ms) {
      const int gmBase = M0 + waveM + ms * 16 + (lane >> 4) * 8;
#pragma unroll
      for (int r = 0; r < 8; ++r) {
        float v = acc[ms][ns][r] + bv;
        v = v > 0.f ? v : 0.f;
        h1[(size_t)(gmBase + r) * HID + gn] = v;
      }
    }
  }
}

// ---------------------------------------------------------------------------
// Kernel 2: out[B_SZ][OUTN] = [x | h1] @ W2^T + b2   (K = 20480)
// grid = (ceil(OUTN/BN)=8, B_SZ/BM), block = 256
// ---------------------------------------------------------------------------
__global__ __launch_bounds__(256) void DPN_gemm2(
    const float* __restrict__ x, const float* __restrict__ h1,
    const float* __restrict__ W2, const float* __restrict__ b2,
    float* __restrict__ out)
{
  __shared__ __align__(16) float As[2][BM * LDT];
  __shared__ __align__(16) float Bs[2][BN * LDT];

  const int tid   = threadIdx.x;
  const int lane  = tid & 31;
  const int wave  = tid >> 5;
  const int lm    = lane & 15;
  const int kh    = (lane >> 4) << 1;
  const int waveM = (wave >> 1) * 32;
  const int waveN = (wave & 1) * 64;

  const int M0 = blockIdx.y * BM;
  const int N0 = blockIdx.x * BN;

  const v8f vzero = {0.f,0.f,0.f,0.f,0.f,0.f,0.f,0.f};
  v8f acc[2][4];
#pragma unroll
  for (int i = 0; i < 2; ++i)
#pragma unroll
    for (int j = 0; j < 4; ++j) acc[i][j] = vzero;

  // Zero the out-of-range W2 rows (n >= 1000) once, in BOTH buffers.
  // Async loads below are predicated off for those rows, so they stay zero.
#pragma unroll
  for (int j = 0; j < 4; ++j) {
    const int idx = tid + j * 256;
    const int row = idx >> 3;
    const int c4  = (idx & 7) << 2;
    if (N0 + row >= OUTN) {
      const float4 z = {0.f, 0.f, 0.f, 0.f};
      *(float4*)(&Bs[0][row * LDT + c4]) = z;
      *(float4*)(&Bs[1][row * LDT + c4]) = z;
    }
  }

  // Issue async loads for K-tile kk into buffer b.
  // A-source switches from x to h1 at k = D_IN (tile-aligned: D_IN % BK == 0).
  auto issue = [&](int kk, int b) {
    const float* Asrc = (kk < D_IN) ? x : h1;
    const int    acol = (kk < D_IN) ? kk : (kk - D_IN);
    const int    alda = (kk < D_IN) ? D_IN : HID;
#pragma unroll
    for (int j = 0; j < 4; ++j) {
      const int idx = tid + j * 256;
      const int row = idx >> 3;
      const int c4  = (idx & 7) << 2;
      async_ld_b128(lds_u32(&As[b][row * LDT + c4]), Asrc,
                    (uint32_t)(((M0 + row) * alda + acol + c4) * 4));
      if (N0 + row < OUTN)
        async_ld_b128(lds_u32(&Bs[b][row * LDT + c4]), W2,
                      (uint32_t)(((N0 + row) * K2TOT + kk + c4) * 4));
    }
  };

  issue(0, 0);
  wait_async0();
  __syncthreads();

  int buf = 0;
  for (int kk = 0; kk < K2TOT; kk += BK) {
    const int nbuf = buf ^ 1;
    if (kk + BK < K2TOT) issue(kk + BK, nbuf);

#pragma unroll
    for (int k4 = 0; k4 < BK; k4 += 4) {
      v2f afrag[2], bfrag[4];
#pragma unroll
      for (int ms = 0; ms < 2; ++ms)
        afrag[ms] = *(const v2f*)(&As[buf][(waveM + ms * 16 + lm) * LDT + k4 + kh]);
#pragma unroll
      for (int ns = 0; ns < 4; ++ns)
        bfrag[ns] = *(const v2f*)(&Bs[buf][(waveN + ns * 16 + lm) * LDT + k4 + kh]);
#pragma unroll
      for (int ms = 0; ms < 2; ++ms)
#pragma unroll
        for (int ns = 0; ns < 4; ++ns)
          acc[ms][ns] = wmma_f32(afrag[ms], bfrag[ns], acc[ms][ns]);
    }

    wait_async0();
    __syncthreads();
    buf = nbuf;
  }

  // Epilogue: bias, bounds-masked store (N = 1000 is not tile-aligned).
#pragma unroll
  for (int ns = 0; ns < 4; ++ns) {
    const int gn  = N0 + waveN + ns * 16 + lm;
    const bool nok = (gn < OUTN);
    const float bv = nok ? b2[gn] : 0.f;
#pragma unroll
    for (int ms = 0; ms < 2; ++ms) {
      const int gmBase = M0 + waveM + ms * 16 + (lane >> 4) * 8;
#pragma unroll
      for (int r = 0; r < 8; ++r) {
        if (nok)
          out[(size_t)(gmBase + r) * OUTN + gn] = acc[ms][ns][r] + bv;
      }
    }
  }
}

// ---------------------------------------------------------------------------
extern "C" void kernel_launch(void* const* d_in, const int* in_sizes, int n_in,
                              void* d_out, int out_size, void* d_ws, size_t ws_size,
                              hipStream_t stream) {
  const float* x  = (const float*)d_in[0];
  const float* W1 = (const float*)d_in[1];
  const float* b1 = (const float*)d_in[2];
  const float* W2 = (const float*)d_in[3];
  const float* b2 = (const float*)d_in[4];
  float* out = (float*)d_out;
  float* h1  = (float*)d_ws;   // [B_SZ][HID] f32 = 256 MiB scratch

  dim3 g1(HID / BN, B_SZ / BM);             // 128 x 32
  DPN_gemm1_relu<<<g1, 256, 0, stream>>>(x, W1, b1, h1);

  dim3 g2((OUTN + BN - 1) / BN, B_SZ / BM); // 8 x 32
  DPN_gemm2<<<g2, 256, 0, stream>>>(x, h1, W2, b2, out);
}